// ReadoutPhase_82686710383217
// MI455X (gfx1250) — compile-verified
//
#include <hip/hip_runtime.h>
#include <math.h>
#include <stdint.h>

#define DIM       256
#define WAVES     8
#define BLOCK     256     // 8 wave32 waves
#define COLS_PER_LANE 8   // 256 cols / 32 lanes
#define PIPE      4       // row buffers per wave (3 rows in flight ahead of consumer)
#define ROW_BYTES (DIM * 4)

// Async DMA of this lane's 8 columns (32 contiguous bytes) of one row into LDS.
// Two b128 async ops (ASYNCcnt += 2 per wave). INST_OFFSET applies to both the
// global and LDS address. Non-temporal: x is 1 GB streamed once -> keep it out
// of the 192 MB L2 so batch/W stay resident across graph replays.
__device__ __forceinline__ void async_copy_cols(const float* gp, unsigned lds_off) {
    asm volatile(
        "global_load_async_to_lds_b128 %0, %1, off th:TH_LOAD_NT\n\t"
        "global_load_async_to_lds_b128 %0, %1, off offset:16 th:TH_LOAD_NT"
        :: "v"(lds_off), "v"(gp)
        : "memory");
}

// Async loads retire in order, so ASYNCcnt <= 2*ahead retires exactly the
// current row while leaving `ahead` prefetched rows in flight.
__device__ __forceinline__ void wait_rows_ahead(int ahead) {
    if (ahead >= 3)      asm volatile("s_wait_asynccnt 6" ::: "memory");
    else if (ahead == 2) asm volatile("s_wait_asynccnt 4" ::: "memory");
    else if (ahead == 1) asm volatile("s_wait_asynccnt 2" ::: "memory");
    else                 asm volatile("s_wait_asynccnt 0" ::: "memory");
}

// Low 32 bits of a flat pointer to __shared__ == LDS byte offset on gfx1250
// (the shared aperture lives in the high 32 bits), i.e. the async VDST value.
__device__ __forceinline__ unsigned lds_byte_off(const void* p) {
    return (unsigned)(uint64_t)(uintptr_t)p;
}

__global__ __launch_bounds__(BLOCK) void readout_gate_seg_kernel(
    const float* __restrict__ x,           // [N, 256]
    const long long* __restrict__ batch,   // [N] sorted segment ids, int64
    const float* __restrict__ W,           // [1, 256]
    const float* __restrict__ bvec,        // [1]
    float* __restrict__ out,               // [NSEG, 512]
    int n_nodes)
{
    // 8 waves x 4 row-buffers x 1KB = 32 KB staging; first half reused for the
    // cross-wave reduction at the end.
    __shared__ __align__(16) float xbuf[WAVES][PIPE][DIM];
    __shared__ int sb[2];

    const int g    = blockIdx.x;
    const int tid  = threadIdx.x;
    const int w    = tid >> 5;    // wave id (wave32)
    const int lane = tid & 31;

    // Segment bounds: lower_bound(batch, g) / lower_bound(batch, g+1).
    if (tid < 2) {
        const long long key = (long long)(g + tid);
        int lo = 0, hi = n_nodes;
        while (lo < hi) {
            int mid = (lo + hi) >> 1;
            if (batch[mid] < key) lo = mid + 1; else hi = mid;
        }
        sb[tid] = lo;
    }
    __syncthreads();
    const int start = sb[0];
    const int end   = sb[1];

    // Per-lane slice of W (8 columns) and bias.
    float wreg[COLS_PER_LANE];
    *(float4*)&wreg[0] = *(const float4*)(W + lane * COLS_PER_LANE);
    *(float4*)&wreg[4] = *(const float4*)(W + lane * COLS_PER_LANE + 4);
    const float bias = bvec[0];

    float acc[COLS_PER_LANE];
    float mx[COLS_PER_LANE];
#pragma unroll
    for (int j = 0; j < COLS_PER_LANE; ++j) { acc[j] = 0.0f; mx[j] = -INFINITY; }

    // This wave's rows: start+w, start+w+8, ...  (all wave-uniform -> EXEC all-1s
    // at every async issue, as the ISA requires for clean VMEM issue).
    const int  r0    = start + w;
    const int  nrows = (end > r0) ? ((end - r0 + WAVES - 1) / WAVES) : 0;
    const float* gp0 = x + (size_t)r0 * DIM + lane * COLS_PER_LANE;
    const unsigned lds_base = lds_byte_off(&xbuf[w][0][lane * COLS_PER_LANE]);

    // Prologue: prime up to PIPE-1 = 3 row buffers.
#pragma unroll
    for (int k = 0; k < PIPE - 1; ++k)
        if (k < nrows)
            async_copy_cols(gp0 + (size_t)k * WAVES * DIM, lds_base + (unsigned)k * ROW_BYTES);

    for (int i = 0; i < nrows; ++i) {
        const int ip = i + (PIPE - 1);
        if (ip < nrows)
            async_copy_cols(gp0 + (size_t)ip * WAVES * DIM,
                            lds_base + (unsigned)(ip & (PIPE - 1)) * ROW_BYTES);

        wait_rows_ahead(nrows - 1 - i);   // retire exactly row i

        // This lane's 8 columns of row i from LDS (2x ds_load_b128).
        const float4* bp = (const float4*)&xbuf[w][i & (PIPE - 1)][lane * COLS_PER_LANE];
        float v[COLS_PER_LANE];
        *(float4*)&v[0] = bp[0];
        *(float4*)&v[4] = bp[1];

        // dot(x_r, W): lane partial + wave32 xor-shuffle tree.
        float p = 0.0f;
#pragma unroll
        for (int j = 0; j < COLS_PER_LANE; ++j) p = fmaf(v[j], wreg[j], p);
#pragma unroll
        for (int off = 16; off >= 1; off >>= 1) p += __shfl_xor(p, off, 32);

        const float score = 1.0f / (1.0f + __expf(-(p + bias)));

#pragma unroll
        for (int j = 0; j < COLS_PER_LANE; ++j) {
            acc[j] = fmaf(score, v[j], acc[j]);
            mx[j]  = fmaxf(mx[j], v[j]);
        }
    }

    asm volatile("s_wait_asynccnt 0" ::: "memory");  // drain before LDS reuse
    __syncthreads();

    // Cross-wave reduction: reuse xbuf as [wave][{sum,max}][col].
#pragma unroll
    for (int j = 0; j < COLS_PER_LANE; ++j) {
        xbuf[w][0][lane * COLS_PER_LANE + j] = acc[j];
        xbuf[w][1][lane * COLS_PER_LANE + j] = mx[j];
    }
    __syncthreads();

    // Thread t owns output column t; write-once output -> non-temporal stores.
    float s = 0.0f, m = -INFINITY;
#pragma unroll
    for (int wv = 0; wv < WAVES; ++wv) {
        s += xbuf[wv][0][tid];
        m  = fmaxf(m, xbuf[wv][1][tid]);
    }
    const size_t ob = (size_t)g * (2 * DIM);
    __builtin_nontemporal_store(s, &out[ob + tid]);        // gated segment-sum
    __builtin_nontemporal_store(m, &out[ob + DIM + tid]);  // segment-max
}

extern "C" void kernel_launch(void* const* d_in, const int* in_sizes, int n_in,
                              void* d_out, int out_size, void* d_ws, size_t ws_size,
                              hipStream_t stream) {
    (void)n_in; (void)d_ws; (void)ws_size;
    const float*     x     = (const float*)d_in[0];
    const long long* batch = (const long long*)d_in[1];  // int64 per reference
    const float*     W     = (const float*)d_in[2];
    const float*     b     = (const float*)d_in[3];
    float*           out   = (float*)d_out;

    const int n_nodes = in_sizes[0] / DIM;
    const int n_seg   = out_size / (2 * DIM);

    readout_gate_seg_kernel<<<dim3(n_seg), dim3(BLOCK), 0, stream>>>(
        x, batch, W, b, out, n_nodes);
}